// InterCBLoss_50139448213776
// MI455X (gfx1250) — compile-verified
//
// InterCBLoss for MI455X (gfx1250): sort-free top-k via 3-level radix select
// over float bit patterns, fused sigmoid/log pass with async-to-LDS double
// buffering (ASYNCcnt pipeline), deterministic fixed-point u64 accumulation.
#include <hip/hip_runtime.h>
#include <stdint.h>

#define NB    16
#define NPIX  (1024 * 1024)
#define EPSV  1e-7f
#define FIXS  67108864.0f            // 2^26 fixed-point scale
#define INV_FIXS (1.0 / 67108864.0)

// ---- workspace layout (bytes) ----
#define OFF_BL   0x0ULL              // float  b_loss[16][NPIX]   (64 MiB)
#define OFF_H1S  0x4000000ULL        // u64    h1_sum[16][2048]
#define OFF_H1C  0x4040000ULL        // u32    h1_cnt[16][2048]
#define OFF_H2S  0x4060000ULL        // u64    h2_sum[16][4096]
#define OFF_H2C  0x40E0000ULL        // u32    h2_cnt[16][4096]
#define OFF_H3S  0x4120000ULL        // u64    h3_sum[16][512]
#define OFF_H3C  0x4130000ULL        // u32    h3_cnt[16][512]
#define OFF_FSUM 0x4138000ULL        // u64[16]
#define OFF_NZ   0x4138080ULL        // u64[16]
#define OFF_BTOT 0x4138100ULL        // u64[16]
#define OFF_ASUM 0x4138180ULL        // u64[16] accumulated above-threshold sum
#define OFF_ACNT 0x4138200ULL        // u64[16] accumulated above-threshold cnt
#define OFF_T1   0x4138280ULL        // u32[16]
#define OFF_T2   0x41382C0ULL        // u32[16]
#define OFF_T3   0x4138300ULL        // u32[16]
#define ZERO_BYTES (0x4138340ULL - 0x4000000ULL)

typedef unsigned long long u64;
typedef unsigned u32;

// ---- CDNA5 async global->LDS streaming (inline asm: portable across
// ROCm7.2 / amdgpu-toolchain builtin arity differences) ----
__device__ __forceinline__ void async_ld_b128(u32 lds_off, u32 goff, const void* sbase) {
  asm volatile("global_load_async_to_lds_b128 %0, %1, %2"
               :: "v"(lds_off), "v"(goff), "s"(sbase) : "memory");
}
__device__ __forceinline__ void wait_async3() { asm volatile("s_wait_asynccnt 0x3" ::: "memory"); }
__device__ __forceinline__ void wait_async0() { asm volatile("s_wait_asynccnt 0x0" ::: "memory"); }

__device__ __forceinline__ u32 lds_off_of(const void* p) {
  // flat LDS aperture: low 32 bits == group-segment byte offset
  return (u32)(uintptr_t)p;
}

__global__ void zero_k(u32* p, int n) {
  int i = blockIdx.x * 256 + threadIdx.x;
  if (i < n) p[i] = 0u;
}

// ---- pass 1: fused loss + level-1 histogram (bits[31:21], 2048 bins) ----
__global__ __launch_bounds__(256) void pass1_k(
    const float* __restrict__ probs, const float* __restrict__ gt,
    float* __restrict__ bl, u64* __restrict__ h1s, u32* __restrict__ h1c,
    u64* __restrict__ fsum, u64* __restrict__ nzcnt) {
  __shared__ u32 hcnt[2048];
  __shared__ u64 hsum[2048];
  __shared__ float stage[2][3][1024];   // double-buffered: ch0, ch1, gt
  __shared__ u64 blk_f;
  __shared__ u32 blk_nz;

  const int t   = threadIdx.x;
  const int s   = blockIdx.x >> 8;      // sample
  const int blk = blockIdx.x & 255;     // 256 blocks/sample, 4096 px each
  const long base = (long)blk * 4096;
  const float* p0 = probs + (size_t)(s * 2 + 0) * NPIX;
  const float* p1 = probs + (size_t)(s * 2 + 1) * NPIX;
  const float* gp = gt + (size_t)s * NPIX;
  float* blp = bl + (size_t)s * NPIX;

  for (int i = t; i < 2048; i += 256) { hcnt[i] = 0u; hsum[i] = 0ull; }
  if (t == 0) { blk_f = 0ull; blk_nz = 0u; }
  __syncthreads();

  auto issue = [&](int tile) {
    const int par = tile & 1;
    const u32 goff = (u32)((base + (long)tile * 1024) * 4) + (u32)t * 16u;
    async_ld_b128(lds_off_of(&stage[par][0][t * 4]), goff, p0);
    async_ld_b128(lds_off_of(&stage[par][1][t * 4]), goff, p1);
    async_ld_b128(lds_off_of(&stage[par][2][t * 4]), goff, gp);
  };

  u64 loc_f = 0ull;
  u32 loc_nz = 0u, loc_zc = 0u;

  auto do1 = [&](float x0, float x1, float gg) -> float {
    float d  = x1 - x0;
    float pc = 1.0f / (1.0f + __expf(-d));        // softmax over 2 ch
    pc = fminf(fmaxf(pc, EPSV), 1.0f - EPSV);
    float f = gg * (-__logf(pc));
    float b = (1.0f - gg) * (-__logf(1.0f - pc));
    if (f != 0.0f) loc_f += (u64)(f * FIXS + 0.5f);
    u32 bits = __float_as_uint(b);
    if (bits != 0u) {
      loc_nz++;
      u32 bin = bits >> 21;
      atomicAdd(&hcnt[bin], 1u);
      atomicAdd(&hsum[bin], (u64)(b * FIXS + 0.5f));
    } else {
      loc_zc++;                                    // batch bin-0 counts
    }
    return b;
  };

  issue(0);
  for (int tile = 0; tile < 4; ++tile) {
    if (tile + 1 < 4) { issue(tile + 1); wait_async3(); }
    else              { wait_async0(); }
    const int par = tile & 1;
    float4 x0 = *(const float4*)&stage[par][0][t * 4];
    float4 x1 = *(const float4*)&stage[par][1][t * 4];
    float4 gv = *(const float4*)&stage[par][2][t * 4];
    float4 bo;
    bo.x = do1(x0.x, x1.x, gv.x);
    bo.y = do1(x0.y, x1.y, gv.y);
    bo.z = do1(x0.z, x1.z, gv.z);
    bo.w = do1(x0.w, x1.w, gv.w);
    *(float4*)&blp[base + (long)tile * 1024 + t * 4] = bo;
  }

  atomicAdd(&blk_f, loc_f);
  atomicAdd(&blk_nz, loc_nz);
  if (loc_zc) atomicAdd(&hcnt[0], loc_zc);
  __syncthreads();

  u64* HS = h1s + (size_t)s * 2048;
  u32* HC = h1c + (size_t)s * 2048;
  for (int i = t; i < 2048; i += 256) {
    u32 c = hcnt[i];
    if (c) atomicAdd(&HC[i], c);
    u64 sm = hsum[i];
    if (sm) atomicAdd(&HS[i], sm);
  }
  if (t == 0) { atomicAdd(&fsum[s], blk_f); atomicAdd(&nzcnt[s], (u64)blk_nz); }
}

// ---- pass 2: refine threshold bin, bits[20:9] (4096 bins) ----
__global__ __launch_bounds__(256) void pass2_k(
    const float* __restrict__ bl, const u32* __restrict__ T1a,
    u64* __restrict__ h2s, u32* __restrict__ h2c) {
  __shared__ u32 hcnt[4096];
  __shared__ u64 hsum[4096];
  const int t = threadIdx.x, s = blockIdx.x >> 8, blk = blockIdx.x & 255;
  const u32 T1 = T1a[s];
  for (int i = t; i < 4096; i += 256) { hcnt[i] = 0u; hsum[i] = 0ull; }
  __syncthreads();
  if (T1 <= 0x7FFu) {
    const float* blp = bl + (size_t)s * NPIX + (size_t)blk * 4096;
    for (int j = 0; j < 4; ++j) {
      float4 v = *(const float4*)&blp[j * 1024 + t * 4];
      float vv[4] = {v.x, v.y, v.z, v.w};
      for (int q = 0; q < 4; ++q) {
        u32 bits = __float_as_uint(vv[q]);
        if ((bits >> 21) == T1) {
          u32 bin = (bits >> 9) & 0xFFFu;
          atomicAdd(&hcnt[bin], 1u);
          if (bits) atomicAdd(&hsum[bin], (u64)(vv[q] * FIXS + 0.5f));
        }
      }
    }
  }
  __syncthreads();
  u64* HS = h2s + (size_t)s * 4096;
  u32* HC = h2c + (size_t)s * 4096;
  for (int i = t; i < 4096; i += 256) {
    u32 c = hcnt[i];
    if (c) atomicAdd(&HC[i], c);
    u64 sm = hsum[i];
    if (sm) atomicAdd(&HS[i], sm);
  }
}

// ---- pass 3: refine again, bits[8:0] (512 bins) ----
__global__ __launch_bounds__(256) void pass3_k(
    const float* __restrict__ bl, const u32* __restrict__ T1a, const u32* __restrict__ T2a,
    u64* __restrict__ h3s, u32* __restrict__ h3c) {
  __shared__ u32 hcnt[512];
  __shared__ u64 hsum[512];
  const int t = threadIdx.x, s = blockIdx.x >> 8, blk = blockIdx.x & 255;
  const u32 T1 = T1a[s], T2 = T2a[s];
  for (int i = t; i < 512; i += 256) { hcnt[i] = 0u; hsum[i] = 0ull; }
  __syncthreads();
  if (T1 <= 0x7FFu && T2 <= 0xFFFu) {
    const u32 key = (T1 << 12) | T2;
    const float* blp = bl + (size_t)s * NPIX + (size_t)blk * 4096;
    for (int j = 0; j < 4; ++j) {
      float4 v = *(const float4*)&blp[j * 1024 + t * 4];
      float vv[4] = {v.x, v.y, v.z, v.w};
      for (int q = 0; q < 4; ++q) {
        u32 bits = __float_as_uint(vv[q]);
        if ((bits >> 9) == key) {
          u32 bin = bits & 0x1FFu;
          atomicAdd(&hcnt[bin], 1u);
          if (bits) atomicAdd(&hsum[bin], (u64)(vv[q] * FIXS + 0.5f));
        }
      }
    }
  }
  __syncthreads();
  u64* HS = h3s + (size_t)s * 512;
  u32* HC = h3c + (size_t)s * 512;
  for (int i = t; i < 512; i += 256) {
    u32 c = hcnt[i];
    if (c) atomicAdd(&HC[i], c);
    u64 sm = hsum[i];
    if (sm) atomicAdd(&HS[i], sm);
  }
}

// ---- select: find threshold bin (scan from top), accumulate strictly-above ----
__global__ __launch_bounds__(256) void select_k(
    const u64* __restrict__ hs, const u32* __restrict__ hc, int nbins,
    const float* __restrict__ gtf, u64* __restrict__ asum, u64* __restrict__ acnt,
    u32* __restrict__ Tout, u64* __restrict__ btot) {
  __shared__ u32 lc[4096];
  __shared__ u64 ls[4096];
  __shared__ u32 sfc[256];
  __shared__ u64 sfs[256];
  __shared__ long long kk_sh;
  __shared__ int done_flag;
  const int t = threadIdx.x, s = blockIdx.x;
  const u64* HS = hs + (size_t)s * nbins;
  const u32* HC = hc + (size_t)s * nbins;
  for (int i = t; i < nbins; i += 256) { lc[i] = HC[i]; ls[i] = HS[i]; }
  __syncthreads();
  const int ch = nbins >> 8;            // bins per thread chunk
  u32 cc = 0; u64 cs = 0;
  for (int j = 0; j < ch; ++j) { cc += lc[t * ch + j]; cs += ls[t * ch + j]; }
  sfc[t] = cc; sfs[t] = cs;
  __syncthreads();
  if (t == 0) {
    u32 rc = 0; u64 rs = 0;
    for (int c = 255; c >= 0; --c) {    // exclusive suffix sums (top-down)
      u32 tc = sfc[c]; u64 ts = sfs[c];
      sfc[c] = rc; sfs[c] = rs;
      rc += tc; rs += ts;
    }
    if (btot) btot[s] = rs;             // level-1 also yields total sum
    long long k  = (long long)gtf[s];
    long long kk = k - (long long)acnt[s];
    kk_sh = kk;
    done_flag = (kk <= 0);
    if (kk <= 0) Tout[s] = 0xFFFFFFFFu; // sentinel: no refinement needed
  }
  __syncthreads();
  if (done_flag) return;
  const long long kk = kk_sh;
  const u32 sc = sfc[t]; const u64 ss = sfs[t];
  if ((long long)sc < kk && (long long)(sc + cc) >= kk) {  // crossing in my chunk
    u32 ac = sc; u64 as = ss; int T = t * ch;
    for (int j = ch - 1; j >= 0; --j) {
      const int b = t * ch + j;
      const u32 c = lc[b];
      if ((long long)(ac + c) >= kk) { T = b; break; }
      ac += c; as += ls[b];
    }
    acnt[s] += (u64)ac;                 // single writer
    asum[s] += as;
    Tout[s] = (u32)T;
  }
}

// ---- finalize: combine per-sample stats into the scalar loss ----
__global__ __launch_bounds__(32) void finalize_k(
    const float* __restrict__ gtf,
    const u64* __restrict__ fsum, const u64* __restrict__ nzc,
    const u64* __restrict__ btot, const u64* __restrict__ asum, const u64* __restrict__ acnt,
    const u32* __restrict__ T1a, const u32* __restrict__ T2a, const u32* __restrict__ T3a,
    float* __restrict__ out) {
  const int t = threadIdx.x;
  double per = 0.0;
  if (t < NB) {
    double k   = (double)(long long)gtf[t];
    double nzv = (double)nzc[t];
    double tot = (double)btot[t] * INV_FIXS;
    double asv = (double)asum[t] * INV_FIXS;
    double acv = (double)acnt[t];
    u32 T1 = T1a[t], T2 = T2a[t], T3 = T3a[t];
    float thv = 0.0f;
    if (T1 <= 0x7FFu && T2 <= 0xFFFu && T3 <= 0x1FFu)
      thv = __uint_as_float((T1 << 21) | (T2 << 9) | T3);
    double need = k - acv; if (need < 0.0) need = 0.0;     // tied-at-threshold picks
    double dsum = asv + need * (double)thv;
    double dnum = fmin(k, nzv);
    double dl = dsum / (dnum + 1e-16);
    double sl = (tot - dsum) / ((double)NPIX + 1e-16);
    double fl = ((double)fsum[t] * INV_FIXS) / ((double)gtf[t] + 1e-16);
    per = dl + sl + fl;
  }
  for (int o = 16; o > 0; o >>= 1) per += __shfl_down(per, o, 32);
  if (t == 0) out[0] = (float)(per / (double)NB);
}

extern "C" void kernel_launch(void* const* d_in, const int* in_sizes, int n_in,
                              void* d_out, int out_size, void* d_ws, size_t ws_size,
                              hipStream_t stream) {
  const float* probs = (const float*)d_in[0];
  const float* gt    = (const float*)d_in[1];
  const float* gtf   = (const float*)d_in[2];
  char* w = (char*)d_ws;

  float* bl = (float*)(w + OFF_BL);
  u64* h1s = (u64*)(w + OFF_H1S);  u32* h1c = (u32*)(w + OFF_H1C);
  u64* h2s = (u64*)(w + OFF_H2S);  u32* h2c = (u32*)(w + OFF_H2C);
  u64* h3s = (u64*)(w + OFF_H3S);  u32* h3c = (u32*)(w + OFF_H3C);
  u64* fsum = (u64*)(w + OFF_FSUM);
  u64* nzc  = (u64*)(w + OFF_NZ);
  u64* btot = (u64*)(w + OFF_BTOT);
  u64* asum = (u64*)(w + OFF_ASUM);
  u64* acnt = (u64*)(w + OFF_ACNT);
  u32* T1 = (u32*)(w + OFF_T1);
  u32* T2 = (u32*)(w + OFF_T2);
  u32* T3 = (u32*)(w + OFF_T3);

  const int zero_words = (int)(ZERO_BYTES / 4);
  zero_k<<<(zero_words + 255) / 256, 256, 0, stream>>>((u32*)(w + OFF_H1S), zero_words);

  const int blocks = NB * 256;
  pass1_k<<<blocks, 256, 0, stream>>>(probs, gt, bl, h1s, h1c, fsum, nzc);
  select_k<<<NB, 256, 0, stream>>>(h1s, h1c, 2048, gtf, asum, acnt, T1, btot);
  pass2_k<<<blocks, 256, 0, stream>>>(bl, T1, h2s, h2c);
  select_k<<<NB, 256, 0, stream>>>(h2s, h2c, 4096, gtf, asum, acnt, T2, nullptr);
  pass3_k<<<blocks, 256, 0, stream>>>(bl, T1, T2, h3s, h3c);
  select_k<<<NB, 256, 0, stream>>>(h3s, h3c, 512, gtf, asum, acnt, T3, nullptr);
  finalize_k<<<1, 32, 0, stream>>>(gtf, fsum, nzc, btot, asum, acnt, T1, T2, T3, (float*)d_out);
}